// DistributedXTW_fourway_66288525247025
// MI455X (gfx1250) — compile-verified
//
#include <hip/hip_runtime.h>

// ---------------------------------------------------------------------------
// out[t][n] = sum_k x[k][t] * W[k][n] + b[n]
//   x: [256, 131072] f32, W: [256, 256] f32, b: [256] f32, out: [131072, 256] f32
//
// MI455X / gfx1250, wave32. fp32-exact path via V_WMMA_F32_16X16X4_F32.
//   * Pre-pass swizzles W (256 KB, L2-resident) into d_ws so each lane gets
//     all four B fragments of a K-step from two coalesced global_load_b128.
//   * Each block covers one 32-token M-slab (grid=4096): a single K-sweep per
//     wave, so nothing is loop-invariant -> no hoisting/spilling.
//   * Double-buffered groups of 4 K-steps pipeline A (HBM stream) and B (L2)
//     loads against the WMMA quads; 8 waves/block hide remaining latency.
// Roofline: 268 MB @ 23.3 TB/s => ~11.5 us floor; ~8.4M f32 WMMAs total.
// ---------------------------------------------------------------------------

typedef float v2f __attribute__((ext_vector_type(2)));
typedef float v4f __attribute__((ext_vector_type(4)));
typedef float v8f __attribute__((ext_vector_type(8)));

#define IN_DIM   256
#define OUT_DIM  256
#define TOKENS   131072

#define MB       32              // tokens per block (2 mslots x 16)
#define NSTEPS   (IN_DIM / 4)    // 64 K-steps of 4
#define G        4               // K-steps per pipeline group
#define NGROUPS  (NSTEPS / G)    // 16 groups

// --------------------------- W reformat pre-pass ---------------------------
// wsw[p*512 + nslot*128 + l*8 + t*2 + parity] = W[(2p+parity)*256 + nslot*64 + 16*t + l]
__global__ __launch_bounds__(256)
void w_swizzle_kernel(const float* __restrict__ W, float* __restrict__ wsw)
{
    const int id    = blockIdx.x * 256 + threadIdx.x;   // 0..8191
    const int l     = id & 15;
    const int nslot = (id >> 4) & 3;
    const int p     = id >> 6;                          // 0..127
    const int n0    = nslot * 64 + l;

    float v[8];
#pragma unroll
    for (int t = 0; t < 4; ++t) {
        v[2 * t]     = W[(2 * p)     * OUT_DIM + n0 + 16 * t];
        v[2 * t + 1] = W[(2 * p + 1) * OUT_DIM + n0 + 16 * t];
    }
    float* dst = wsw + p * 512 + nslot * 128 + l * 8;
    ((v4f*)dst)[0] = (v4f){v[0], v[1], v[2], v[3]};
    ((v4f*)dst)[1] = (v4f){v[4], v[5], v[6], v[7]};
}

// ------------------------------- main GEMM ---------------------------------
__global__ __launch_bounds__(256)
void xtw_wmma_f32_kernel(const float* __restrict__ x,
                         const float* __restrict__ wsw,
                         const float* __restrict__ bias_v,
                         float* __restrict__ out)
{
    const int tid   = threadIdx.x;
    const int lane  = tid & 31;                   // wave32
    const int wave  = tid >> 5;                   // 8 waves
    const int half  = lane >> 4;                  // K-pair select per ISA A/B layout
    const int l     = lane & 15;

    const int nslot = wave & 3;                   // 4 slots x 64 output columns
    const int mslot = wave >> 2;                  // 2 slots x 16 tokens

    const int ncol0  = nslot * 64 + l;
    const int tblock = blockIdx.x * MB;           // this block's 32-token slab

    // Per-lane base into swizzled W: two b128 loads per K-step = all 4 B frags.
    const v4f* wq = (const v4f*)wsw + (size_t)half * 128 + nslot * 32 + l * 2;

    // Per-lane base into x: A fragment for K-step s is xa[(4s+2*half+{0,1})*TOKENS].
    const int tA = tblock + mslot * 16 + l;
    const float* xa = x + tA;

    float bv[4];
#pragma unroll
    for (int t = 0; t < 4; ++t) bv[t] = bias_v[ncol0 + 16 * t];

    v8f acc0 = {}, acc1 = {}, acc2 = {}, acc3 = {};

    v2f a[2][G];
    v4f q[2][G][2];

    // ---- prologue: load group 0 ----
#pragma unroll
    for (int i = 0; i < G; ++i) {
        a[0][i].x = xa[(size_t)(4 * i + 2 * half)     * TOKENS];
        a[0][i].y = xa[(size_t)(4 * i + 2 * half + 1) * TOKENS];
        q[0][i][0] = wq[(size_t)i * 256];         // 4096 B per K-step
        q[0][i][1] = wq[(size_t)i * 256 + 1];
    }

#pragma unroll
    for (int g = 0; g < NGROUPS; ++g) {
        const int cur = g & 1;
        // ---- issue loads for group g+1 into the other buffer ----
        if (g + 1 < NGROUPS) {
            const int nxt = (g + 1) & 1;
#pragma unroll
            for (int i = 0; i < G; ++i) {
                const int s = (g + 1) * G + i;
                a[nxt][i].x = xa[(size_t)(4 * s + 2 * half)     * TOKENS];
                a[nxt][i].y = xa[(size_t)(4 * s + 2 * half + 1) * TOKENS];
                q[nxt][i][0] = wq[(size_t)s * 256];
                q[nxt][i][1] = wq[(size_t)s * 256 + 1];
            }
        }
        // ---- compute group g: 4 K-steps x 4 N-tiles ----
#pragma unroll
        for (int i = 0; i < G; ++i) {
            const v2f av = a[cur][i];
            const v2f b0 = q[cur][i][0].xy;
            const v2f b1 = q[cur][i][0].zw;
            const v2f b2 = q[cur][i][1].xy;
            const v2f b3 = q[cur][i][1].zw;
            acc0 = __builtin_amdgcn_wmma_f32_16x16x4_f32(false, av, false, b0, (short)0, acc0, false, false);
            acc1 = __builtin_amdgcn_wmma_f32_16x16x4_f32(false, av, false, b1, (short)0, acc1, false, false);
            acc2 = __builtin_amdgcn_wmma_f32_16x16x4_f32(false, av, false, b2, (short)0, acc2, false, false);
            acc3 = __builtin_amdgcn_wmma_f32_16x16x4_f32(false, av, false, b3, (short)0, acc3, false, false);
        }
    }

    // ---- bias add + store. C layout: VGPR v -> M = v + 8*half, N = lane%16 ----
    const int trow0 = tblock + mslot * 16 + 8 * half;
#pragma unroll
    for (int v = 0; v < 8; ++v) {
        float* orow = out + (size_t)(trow0 + v) * OUT_DIM;
        orow[ncol0 +  0] = acc0[v] + bv[0];
        orow[ncol0 + 16] = acc1[v] + bv[1];
        orow[ncol0 + 32] = acc2[v] + bv[2];
        orow[ncol0 + 48] = acc3[v] + bv[3];
    }
}

extern "C" void kernel_launch(void* const* d_in, const int* in_sizes, int n_in,
                              void* d_out, int out_size, void* d_ws, size_t ws_size,
                              hipStream_t stream) {
    const float* x = (const float*)d_in[0];   // [256, 131072]
    const float* W = (const float*)d_in[1];   // [256, 256]
    const float* b = (const float*)d_in[2];   // [256]
    // d_in[3] = model_group handle: unused on single device.
    float* out = (float*)d_out;               // [131072, 256]
    float* wsw = (float*)d_ws;                // needs 256 KiB of scratch

    // Pre-pass: swizzle W into d_ws (8192 threads, one per (p, nslot, lane)).
    hipLaunchKernelGGL(w_swizzle_kernel, dim3(32), dim3(256), 0, stream, W, wsw);

    // Main GEMM: 4096 blocks x 8 waves, 32 tokens per block, single K-sweep.
    hipLaunchKernelGGL(xtw_wmma_f32_kernel, dim3(TOKENS / MB), dim3(256),
                       0, stream, x, wsw, b, out);
}